// GCN_7181185319266
// MI455X (gfx1250) — compile-verified
//
#include <hip/hip_runtime.h>

typedef __attribute__((ext_vector_type(2))) float v2f;
typedef __attribute__((ext_vector_type(4))) float v4f;
typedef __attribute__((ext_vector_type(8))) float v8f;

// -----------------------------------------------------------------------------
// Dense GEMM with bias:  C[M,N] = op(A)[M,K] @ B[K,N] + bias[N]
// op(A) = relu(A) if RELU_A (fuses the GCN relu into the 2nd layer's A-load).
// One wave computes one 16x16 tile via V_WMMA_F32_16X16X4_F32 (exact fp32).
// Block = 256 threads = 8 waves; NT_M m-tiles x NT_N n-tiles per block.
//
// Fragment layouts (CDNA5 ISA 7.12.2, wave32):
//   A 16x4 : lane L<16 -> row M=L,  v0=K0 v1=K1 ; lane L>=16 -> row M=L-16, v0=K2 v1=K3
//   B 4x16 : lane L<16 -> col N=L,  v0=K0 v1=K1 ; lane L>=16 -> col N=L-16, v0=K2 v1=K3
//   C/D    : VGPR j: lanes 0-15 -> M=j, N=lane ; lanes 16-31 -> M=j+8, N=lane-16
// -----------------------------------------------------------------------------
template <int K, int N, int NT_M, int NT_N, bool RELU_A>
__global__ __launch_bounds__(256) void gemm_bias_wmma(
    const float* __restrict__ A, const float* __restrict__ B,
    const float* __restrict__ bias, float* __restrict__ C) {
  const int lane = threadIdx.x & 31;
  const int wave = threadIdx.x >> 5;
  const int mt   = wave / NT_N;
  const int nt   = wave % NT_N;
  const int row0 = (blockIdx.x * NT_M + mt) * 16;
  const int col0 = nt * 16;
  const int lh   = lane & 15;  // 0..15
  const int hi   = lane >> 4;  // 0 or 1 (selects K pair 2,3)

  const float* Ap = A + (row0 + lh) * K + hi * 2;  // advance +4 per step
  const float* Bp = B + (hi * 2) * N + col0 + lh;  // advance +4*N per step

  v8f acc;
  {
    const float bv = bias[col0 + lh];
#pragma unroll
    for (int j = 0; j < 8; ++j) acc[j] = bv;
  }

#pragma unroll 4
  for (int k = 0; k < K; k += 4) {
    v2f a;
    a.x = Ap[0];
    a.y = Ap[1];
    if (RELU_A) {
      a.x = fmaxf(a.x, 0.0f);
      a.y = fmaxf(a.y, 0.0f);
    }
    v2f b;
    b.x = Bp[0];
    b.y = Bp[N];
    acc = __builtin_amdgcn_wmma_f32_16x16x4_f32(
        /*neg_a=*/false, a, /*neg_b=*/false, b,
        /*c_mod=*/(short)0, acc, /*reuse_a=*/false, /*reuse_b=*/false);
    Ap += 4;
    Bp += 4 * N;
  }

  float* Cp = C + (row0 + hi * 8) * N + col0 + lh;
#pragma unroll
  for (int j = 0; j < 8; ++j) Cp[j * N] = acc[j];
}

// -----------------------------------------------------------------------------
// SpMM scatter-add: out[row[e], f] += val[e] * x[col[e], f]
// One thread per (edge, 4 features): b128 gather of x's row segment, edge
// metadata amortized 4x, then 4 fp32 atomics. F/4 consecutive threads cover
// one edge -> coalesced row gather; atomics land in L2 (h and out both fit
// in MI455X's 192MB L2).
// -----------------------------------------------------------------------------
template <int LOGF>
__global__ __launch_bounds__(256) void spmm_scatter4(
    const int* __restrict__ row, const int* __restrict__ col,
    const float* __restrict__ val, const float* __restrict__ x,
    float* __restrict__ out, int nWork) {
  const int idx = blockIdx.x * 256 + threadIdx.x;
  if (idx >= nWork) return;
  constexpr int LOGQ = LOGF - 2;        // quads per row
  const int e = idx >> LOGQ;
  const int q = idx & ((1 << LOGQ) - 1);
  const int r = row[e];
  const int c = col[e];
  const float v = val[e];

  const v4f* __restrict__ x4 = (const v4f*)x;
  const v4f g = x4[((size_t)c << LOGQ) + q];  // global_load_b128

  float* o = out + ((size_t)r << LOGF) + (q << 2);
  unsafeAtomicAdd(o + 0, v * g.x);
  unsafeAtomicAdd(o + 1, v * g.y);
  unsafeAtomicAdd(o + 2, v * g.z);
  unsafeAtomicAdd(o + 3, v * g.w);
}

extern "C" void kernel_launch(void* const* d_in, const int* in_sizes, int n_in,
                              void* d_out, int out_size, void* d_ws,
                              size_t ws_size, hipStream_t stream) {
  const float* x       = (const float*)d_in[0];
  const int*   adj_row = (const int*)d_in[1];
  const int*   adj_col = (const int*)d_in[2];
  const float* adj_val = (const float*)d_in[3];
  const float* w1      = (const float*)d_in[4];
  const float* b1      = (const float*)d_in[5];
  const float* w2      = (const float*)d_in[6];
  const float* b2      = (const float*)d_in[7];
  float*       out     = (float*)d_out;

  constexpr int N_NODES = 100000;
  constexpr int IN_F = 512, HID_F = 128, N_CLS = 64;
  const int n_edges = in_sizes[1];

  char*  ws = (char*)d_ws;
  float* h0 = (float*)ws;                                  // [N, 128]
  float* h  = (float*)(ws + (size_t)N_NODES * HID_F * 4);  // [N, 128]
  float* h2 = h0;  // h0 dead after spmm1 -> reuse for [N, 64]

  // Zero scatter targets (memset nodes are graph-capturable).
  hipMemsetAsync(h, 0, (size_t)N_NODES * HID_F * sizeof(float), stream);
  hipMemsetAsync(out, 0, (size_t)N_NODES * N_CLS * sizeof(float), stream);

  // Layer 1 projection: h0 = x @ w1 + b1   (M=100000 = 6250 * 16)
  gemm_bias_wmma<IN_F, HID_F, 1, 8, false>
      <<<N_NODES / 16, 256, 0, stream>>>(x, w1, b1, h0);

  // Layer 1 aggregation: h += scatter(val * h0[col])
  {
    const int nWork = n_edges * (HID_F / 4);  // 51.2M threads
    spmm_scatter4<7><<<(nWork + 255) / 256, 256, 0, stream>>>(
        adj_row, adj_col, adj_val, h0, h, nWork);
  }

  // Layer 2 projection (relu fused into A-load): h2 = relu(h) @ w2 + b2
  gemm_bias_wmma<HID_F, N_CLS, 2, 4, true>
      <<<N_NODES / 32, 256, 0, stream>>>(h, w2, b2, h2);

  // Layer 2 aggregation into final output.
  {
    const int nWork = n_edges * (N_CLS / 4);  // 25.6M threads
    spmm_scatter4<6><<<(nWork + 255) / 256, 256, 0, stream>>>(
        adj_row, adj_col, adj_val, h2, out, nWork);
  }
}